// SACLRAll_90452011254158
// MI455X (gfx1250) — compile-verified
//
#include <hip/hip_runtime.h>
#include <hip/hip_bf16.h>
#include <math.h>

typedef __attribute__((ext_vector_type(2))) float v2f;
typedef __attribute__((ext_vector_type(8))) float v8f;

#define BATCH   4096
#define DIM     128
#define NTILES  (BATCH / 16)   // 256
#define N_DATA  1281167

// ---------------------------------------------------------------------------
// Kernel 1: rowwise L2 normalization + squared-norms of the normalized rows.
// One wave (32 lanes) per row; each lane handles a float4 (128 = 32*4).
// ---------------------------------------------------------------------------
__global__ void __launch_bounds__(256)
saclr_normalize_kernel(const float* __restrict__ a, const float* __restrict__ b,
                       float* __restrict__ An, float* __restrict__ Bn,
                       float* __restrict__ naA, float* __restrict__ naB)
{
    const int w    = threadIdx.x >> 5;
    const int lane = threadIdx.x & 31;
    const int row  = blockIdx.x * 8 + w;

    const float* src = blockIdx.y ? b  : a;
    float*       dst = blockIdx.y ? Bn : An;
    float*       na  = blockIdx.y ? naB : naA;

    const float4 v = *reinterpret_cast<const float4*>(src + (size_t)row * DIM + lane * 4);
    float s = v.x * v.x + v.y * v.y + v.z * v.z + v.w * v.w;
    #pragma unroll
    for (int m = 1; m <= 16; m <<= 1) s += __shfl_xor(s, m, 32);

    const float n = sqrtf(s);
    float4 o;
    o.x = v.x / n; o.y = v.y / n; o.z = v.z / n; o.w = v.w / n;
    *reinterpret_cast<float4*>(dst + (size_t)row * DIM + lane * 4) = o;

    // norm of the *normalized* row (matches reference's sum(x*x) on normalized x)
    float s2 = o.x * o.x + o.y * o.y + o.z * o.z + o.w * o.w;
    #pragma unroll
    for (int m = 1; m <= 16; m <<= 1) s2 += __shfl_xor(s2, m, 32);
    if (lane == 0) na[row] = s2;
}

// ---------------------------------------------------------------------------
// Kernel 2: tiled Gram matrix via V_WMMA_F32_16X16X4_F32, fused exp + row-sum.
// blockIdx.y selects the product: 0=A.A^T 1=B.B^T 2=A.B^T(+diag pos) 3=B.A^T.
// blockIdx.x = 16-row strip. 8 waves/block, wave strides the 256 N-tiles.
// Fully deterministic: each row's sum is written exactly once by its block.
// ---------------------------------------------------------------------------
__global__ void __launch_bounds__(256)
saclr_gram_exp_kernel(const float* __restrict__ An, const float* __restrict__ Bn,
                      const float* __restrict__ naA, const float* __restrict__ naB,
                      float* __restrict__ rs_aa, float* __restrict__ rs_bb,
                      float* __restrict__ rs_ab, float* __restrict__ rs_ba,
                      float* __restrict__ pos)
{
    const int tile_m = blockIdx.x;
    const float *X, *Y, *nX, *nY;
    float* out;
    bool rec = false;
    switch (blockIdx.y) {
        case 0:  X = An; Y = An; nX = naA; nY = naA; out = rs_aa; break;
        case 1:  X = Bn; Y = Bn; nX = naB; nY = naB; out = rs_bb; break;
        case 2:  X = An; Y = Bn; nX = naA; nY = naB; out = rs_ab; rec = true; break;
        default: X = Bn; Y = An; nX = naB; nY = naA; out = rs_ba; break;
    }

    const int tid  = threadIdx.x;
    const int w    = tid >> 5;
    const int lane = tid & 31;
    const int nlo  = lane & 15;      // row idx (A frag) / col idx (B frag, C/D)
    const int hi   = lane >> 4;      // half select
    const int koff = hi << 1;        // K offset within a K=4 step

    // Preload all A fragments for this 16-row strip: 32 steps of K=4.
    v2f afrag[32];
    const float* xrow = X + (size_t)(tile_m * 16 + nlo) * DIM + koff;
    #pragma unroll
    for (int k = 0; k < 32; ++k)
        afrag[k] = *reinterpret_cast<const v2f*>(xrow + 4 * k);

    float narow[8];
    #pragma unroll
    for (int r = 0; r < 8; ++r)
        narow[r] = nX[tile_m * 16 + r + 8 * hi];

    float racc[8];
    #pragma unroll
    for (int r = 0; r < 8; ++r) racc[r] = 0.0f;

    for (int nt = w; nt < NTILES; nt += 8) {
        const float* yrow = Y + (size_t)(nt * 16 + nlo) * DIM + koff;
        v8f c = {};
        #pragma unroll
        for (int k = 0; k < 32; ++k) {
            v2f bfrag = *reinterpret_cast<const v2f*>(yrow + 4 * k);
            c = __builtin_amdgcn_wmma_f32_16x16x4_f32(
                    /*neg_a=*/false, afrag[k], /*neg_b=*/false, bfrag,
                    /*c_mod=*/(short)0, c, /*reuse_a=*/false, /*reuse_b=*/false);
        }
        const int   gcol  = nt * 16 + nlo;
        const float nacol = nY[gcol];
        #pragma unroll
        for (int r = 0; r < 8; ++r) {
            const int grow = tile_m * 16 + r + 8 * hi;
            const float sq = fmaxf(narow[r] + nacol - 2.0f * c[r], 0.0f);
            const float lg = -2.0f * sq;       // * 1/(2*TEMP^2) with TEMP=0.5
            if (grow == gcol) {
                if (rec) pos[grow] = lg;       // diag masked: exp(lg-1e9)==0
            } else {
                racc[r] += expf(lg);
            }
        }
    }

    // Reduce over the 16 lanes of each half (columns of the C tile).
    #pragma unroll
    for (int m = 1; m <= 8; m <<= 1) {
        #pragma unroll
        for (int r = 0; r < 8; ++r)
            racc[r] += __shfl_xor(racc[r], m, 32);
    }

    __shared__ float lred[8][16];
    if (nlo == 0) {
        #pragma unroll
        for (int r = 0; r < 8; ++r) lred[w][r + 8 * hi] = racc[r];
    }
    __syncthreads();
    if (tid < 16) {
        float t = 0.0f;
        #pragma unroll
        for (int ww = 0; ww < 8; ++ww) t += lred[ww][tid];
        out[tile_m * 16 + tid] = t;
    }
}

// ---------------------------------------------------------------------------
// Kernel 3: scalar reduction to the final loss.
// ---------------------------------------------------------------------------
__global__ void __launch_bounds__(256)
saclr_finalize_kernel(const float* __restrict__ pos,
                      const float* __restrict__ rs_aa, const float* __restrict__ rs_bb,
                      const float* __restrict__ rs_ab, const float* __restrict__ rs_ba,
                      const int* __restrict__ idx, const float* __restrict__ s_inv,
                      float* __restrict__ out)
{
    const float n2    = (float)N_DATA * (float)N_DATA;
    const float scale = 2.0f * (float)BATCH - 2.0f;

    float attr = 0.0f, repa = 0.0f, repb = 0.0f;
    for (int i = threadIdx.x; i < BATCH; i += 256) {
        const float p = pos[i];
        attr += -logf(expf(p));                       // attractive (a == b variant)
        const float z = s_inv[idx[i]] / n2 * scale;   // Z_hat
        repa += (rs_ab[i] + rs_aa[i]) / z;
        repb += (rs_ba[i] + rs_bb[i]) / z;
    }

    __shared__ float sa[256], sra[256], srb[256];
    sa[threadIdx.x] = attr; sra[threadIdx.x] = repa; srb[threadIdx.x] = repb;
    __syncthreads();
    for (int off = 128; off > 0; off >>= 1) {
        if (threadIdx.x < off) {
            sa[threadIdx.x]  += sa[threadIdx.x + off];
            sra[threadIdx.x] += sra[threadIdx.x + off];
            srb[threadIdx.x] += srb[threadIdx.x + off];
        }
        __syncthreads();
    }
    if (threadIdx.x == 0) {
        const float invB = 1.0f / (float)BATCH;
        // loss = attr.mean + (rep_a.mean + rep_b.mean)/2  (attr_a == attr_b)
        out[0] = sa[0] * invB + 0.5f * (sra[0] * invB + srb[0] * invB);
    }
}

// ---------------------------------------------------------------------------
extern "C" void kernel_launch(void* const* d_in, const int* in_sizes, int n_in,
                              void* d_out, int out_size, void* d_ws, size_t ws_size,
                              hipStream_t stream) {
    const float* feats_a = (const float*)d_in[0];
    const float* feats_b = (const float*)d_in[1];
    const int*   fidx    = (const int*)d_in[2];
    const float* s_inv   = (const float*)d_in[3];
    float*       out     = (float*)d_out;

    float* ws = (float*)d_ws;
    float* An    = ws;                       // 4096*128
    float* Bn    = An  + BATCH * DIM;        // 4096*128
    float* naA   = Bn  + BATCH * DIM;        // 4096
    float* naB   = naA + BATCH;              // 4096
    float* pos   = naB + BATCH;              // 4096
    float* rs_aa = pos   + BATCH;            // 4096
    float* rs_bb = rs_aa + BATCH;            // 4096
    float* rs_ab = rs_bb + BATCH;            // 4096
    float* rs_ba = rs_ab + BATCH;            // 4096
    (void)in_sizes; (void)n_in; (void)out_size; (void)ws_size;

    saclr_normalize_kernel<<<dim3(BATCH / 8, 2), 256, 0, stream>>>(
        feats_a, feats_b, An, Bn, naA, naB);

    saclr_gram_exp_kernel<<<dim3(NTILES, 4), 256, 0, stream>>>(
        An, Bn, naA, naB, rs_aa, rs_bb, rs_ab, rs_ba, pos);

    saclr_finalize_kernel<<<1, 256, 0, stream>>>(
        pos, rs_aa, rs_bb, rs_ab, rs_ba, fidx, s_inv, out);
}